// TrackBase_7370163880501
// MI455X (gfx1250) — compile-verified
//
#include <hip/hip_runtime.h>

// ---------------------------------------------------------------------------
// Problem constants (fixed by the reference): B=2, C=256, H=W=360, N1=N2=1024
// ---------------------------------------------------------------------------
#define NC    2048          // total centers (1024 frame0 + 1024 frame1)
#define CH    256           // feature channels
#define IMG_H 360
#define IMG_W 360
#define NPTS  1024

typedef __attribute__((ext_vector_type(2))) float v2f;
typedef __attribute__((ext_vector_type(8))) float v8f;

// ---------------------------------------------------------------------------
// Stage 0: irregular gather  X[i, c] = feature[b, c, h, w]
// ---------------------------------------------------------------------------
__global__ __launch_bounds__(256) void gather_kernel(
    const float* __restrict__ feature,
    const int* __restrict__ idx1,   // [3, 1024] rows: batch, h, w
    const int* __restrict__ idx2,   // [3, 1024]
    float* __restrict__ X)          // [2048, 256]
{
    const int i = blockIdx.x;       // center 0..2047
    const int c = threadIdx.x;      // channel 0..255
    const int* idx = (i < NPTS) ? idx1 : idx2;
    const int j = (i < NPTS) ? i : (i - NPTS);
    const int b = idx[0 * NPTS + j];
    const int h = idx[1 * NPTS + j];
    const int w = idx[2 * NPTS + j];
    X[(size_t)i * CH + c] =
        feature[(((size_t)b * CH + c) * IMG_H + h) * IMG_W + w];
}

// ---------------------------------------------------------------------------
// Stage 1-3: fp32 WMMA GEMM.  C[M,N] = act(A[M,K] @ B[K,N] + bias)
// One wave (32 lanes) per 16x16 output tile; V_WMMA_F32_16X16X4_F32, K step 4.
// A 16x4 layout : lanes 0-15 -> M=lane,  K={0,1}; lanes 16-31 -> M=lane-16, K={2,3}
// B 4x16 layout : lanes 0-15 -> N=lane,  K={0,1}; lanes 16-31 -> N=lane-16, K={2,3}
// C/D          : VGPR r -> row r + 8*(lane>=16), col = lane&15
// ---------------------------------------------------------------------------
__global__ __launch_bounds__(32) void gemm_wmma_f32(
    const float* __restrict__ A, const float* __restrict__ B,
    const float* __restrict__ bias, float* __restrict__ C,
    int M, int N, int K, int do_relu)
{
    const int lane = threadIdx.x & 31;
    const int half = lane >> 4;        // 0: lanes 0-15, 1: lanes 16-31
    const int lm   = lane & 15;
    const int n0 = blockIdx.x * 16;
    const int m0 = blockIdx.y * 16;

    const int m = m0 + lm;             // A row held by this lane
    const int n = n0 + lm;             // B col held by this lane

    v8f acc = {};
    for (int k = 0; k < K; k += 4) {
        const int ka = k + half * 2;
        v2f a, b;
        const float2 av = *(const float2*)(A + (size_t)m * K + ka);  // 8B aligned
        a.x = av.x; a.y = av.y;
        b.x = B[(size_t)ka * N + n];
        b.y = B[(size_t)(ka + 1) * N + n];
        acc = __builtin_amdgcn_wmma_f32_16x16x4_f32(
            /*neg_a=*/false, a, /*neg_b=*/false, b,
            /*c_mod=*/(short)0, acc, /*reuse_a=*/false, /*reuse_b=*/false);
    }

    const float bv = bias ? bias[n] : 0.0f;
#pragma unroll
    for (int r = 0; r < 8; ++r) {
        float v = acc[r] + bv;
        if (do_relu) v = fmaxf(v, 0.0f);
        C[(size_t)(m0 + half * 8 + r) * N + n] = v;
    }
}

// ---------------------------------------------------------------------------
// Stage 4: pairwise kernel exploiting linearity of w3:
//   out[m,n] = b4 + sum_k relu(g1[m,k] + (b3[k] - g2[n,k])) * w4[k]
// G = [2048, 32]: rows 0..1023 are g1 = f1@w3, rows 1024..2047 are g2 = f2@w3.
// 16x16 output tile per block; LDS rows padded to 33 to kill bank conflicts.
// ---------------------------------------------------------------------------
__global__ __launch_bounds__(256) void pairwise_kernel(
    const float* __restrict__ G,
    const float* __restrict__ b3,   // [32]
    const float* __restrict__ w4,   // [32]
    const float* __restrict__ b4,   // [1]
    float* __restrict__ out)        // [1024, 1024]
{
    __shared__ float sg1[16][33];
    __shared__ float sg2[16][33];
    __shared__ float sw4[32];

    const int tx = threadIdx.x;     // n within tile
    const int ty = threadIdx.y;     // m within tile
    const int tid = ty * 16 + tx;
    const int n0 = blockIdx.x * 16;
    const int m0 = blockIdx.y * 16;

    for (int t = tid; t < 512; t += 256) {
        const int r = t >> 5, k = t & 31;
        sg1[r][k] = G[(size_t)(m0 + r) * 32 + k];
        sg2[r][k] = b3[k] - G[(size_t)(NPTS + n0 + r) * 32 + k];
    }
    if (tid < 32) sw4[tid] = w4[tid];
    __syncthreads();

    float acc = 0.0f;
#pragma unroll
    for (int k = 0; k < 32; ++k) {
        const float v = fmaxf(sg1[ty][k] + sg2[tx][k], 0.0f);
        acc = fmaf(v, sw4[k], acc);
    }
    out[(size_t)(m0 + ty) * NPTS + (n0 + tx)] = acc + b4[0];
}

// ---------------------------------------------------------------------------
// Launch
// ---------------------------------------------------------------------------
extern "C" void kernel_launch(void* const* d_in, const int* in_sizes, int n_in,
                              void* d_out, int out_size, void* d_ws, size_t ws_size,
                              hipStream_t stream) {
    const float* feature = (const float*)d_in[0];
    const int*   idx1    = (const int*)d_in[1];
    const int*   idx2    = (const int*)d_in[2];
    const float* w1      = (const float*)d_in[3];
    const float* b1      = (const float*)d_in[4];
    const float* w2      = (const float*)d_in[5];
    const float* b2      = (const float*)d_in[6];
    const float* w3      = (const float*)d_in[7];
    const float* b3      = (const float*)d_in[8];
    const float* w4      = (const float*)d_in[9];
    const float* b4      = (const float*)d_in[10];
    float* out = (float*)d_out;

    // Workspace layout (floats): X[2048*256] | H1[2048*256] | F[2048*128] | G[2048*32]
    float* X  = (float*)d_ws;
    float* H1 = X  + (size_t)NC * 256;
    float* F  = H1 + (size_t)NC * 256;
    float* G  = F  + (size_t)NC * 128;

    // Stage 0: gather per-center feature vectors
    gather_kernel<<<dim3(NC), dim3(256), 0, stream>>>(feature, idx1, idx2, X);

    // Stage 1: H1 = relu(X @ w1 + b1)       [2048,256] x [256,256]
    gemm_wmma_f32<<<dim3(256 / 16, NC / 16), dim3(32), 0, stream>>>(
        X, w1, b1, H1, NC, 256, 256, 1);

    // Stage 2: F = relu(H1 @ w2 + b2)       [2048,256] x [256,128]
    gemm_wmma_f32<<<dim3(128 / 16, NC / 16), dim3(32), 0, stream>>>(
        H1, w2, b2, F, NC, 128, 256, 1);

    // Stage 3: G = F @ w3 (no bias/relu; b3 folded into pairwise stage)
    gemm_wmma_f32<<<dim3(32 / 16, NC / 16), dim3(32), 0, stream>>>(
        F, w3, nullptr, G, NC, 32, 128, 0);

    // Stage 4: out[m,n] = b4 + sum_k relu(g1[m,k] - g2[n,k] + b3[k]) * w4[k]
    pairwise_kernel<<<dim3(NPTS / 16, NPTS / 16), dim3(16, 16), 0, stream>>>(
        G, b3, w4, b4, out);
}